// TXCDRBasisExpansion_52201032516286
// MI455X (gfx1250) — compile-verified
//
#include <hip/hip_runtime.h>
#include <hip/hip_bf16.h>

typedef __attribute__((ext_vector_type(2))) float     v2f;
typedef __attribute__((ext_vector_type(8))) float     v8f;
typedef __attribute__((ext_vector_type(4))) unsigned  v4u;
typedef __attribute__((ext_vector_type(4))) int       v4i;
typedef __attribute__((ext_vector_type(8))) int       v8i;

#define D_IN    768
#define D_SAE   16384
#define T_STEPS 12
#define TOPK    64
#define KBASIS  3
#define BATCH   1024
#define KTOT    (T_STEPS * D_IN)   // 9216
#define NIT     (KTOT / 32)        // 288 k-tiles

#if defined(__has_builtin)
#if __has_builtin(__builtin_amdgcn_tensor_load_to_lds) && __has_builtin(__builtin_amdgcn_s_wait_tensorcnt)
#define HAVE_TDM 1
#endif
#if __has_builtin(__builtin_amdgcn_global_load_async_to_lds_b128) && __has_builtin(__builtin_amdgcn_s_wait_asynccnt)
#define HAVE_ASYNC 1
#endif
#endif
#ifndef HAVE_TDM
#define HAVE_TDM 0
#endif
#ifndef HAVE_ASYNC
#define HAVE_ASYNC 0
#endif

// amdgpu-toolchain (therock headers) ships the TDM header and uses the 6-arg
// builtin; ROCm 7.2 has no header and a 5-arg builtin.
#if __has_include(<hip/amd_detail/amd_gfx1250_TDM.h>)
#define TDM_SIX_ARGS 1
#else
#define TDM_SIX_ARGS 0
#endif

// --- report the selected staging tier in compile stderr (compile-only loop!)
#if HAVE_TDM
#if TDM_SIX_ARGS
#pragma message("CDNA5 staging tier: TENSOR_LOAD_TO_LDS (TDM, 6-arg builtin)")
#else
#pragma message("CDNA5 staging tier: TENSOR_LOAD_TO_LDS (TDM, 5-arg builtin)")
#endif
#elif HAVE_ASYNC
#pragma message("CDNA5 staging tier: GLOBAL_LOAD_ASYNC_TO_LDS_B128 (ASYNCcnt)")
#else
#pragma message("CDNA5 staging tier: plain global loads + ds_store (no TDM/async builtins)")
#endif

typedef const __attribute__((address_space(1))) void* gas_ptr;
typedef __attribute__((address_space(3))) void*       las_ptr;

// ---------------------------------------------------------------------------
// Kernel 1: encode GEMM  pre[b,s] = sum_k x[b,k] * W[k,s] + b_enc[s]
// Block: 256 threads (8 waves). Block tile: 64 rows x 128 cols.
// Double-buffered LDS; W tiles staged by the Tensor Data Mover (preferred;
// in-flight LDS row padding 128->144 dwords) or async global->LDS copies;
// A tiles by plain loads (needs a transpose the DMA paths cannot do).
// ---------------------------------------------------------------------------
__global__ __launch_bounds__(256) void encode_gemm_kernel(
    const float* __restrict__ X,      // (B, KTOT)
    const float* __restrict__ W,      // (KTOT, D_SAE)
    const float* __restrict__ b_enc,  // (D_SAE)
    float* __restrict__ pre)          // (B, D_SAE)
{
    __shared__ float Al[2][32][65];   // A tile transposed [k][m], padded
    __shared__ float Bl[2][32][144];  // B tile [k][n], padded (288 % 64 = 32)

    const int tid  = threadIdx.x;
    const int wave = tid >> 5;
    const int lane = tid & 31;
    const int half = lane >> 4;       // 0 or 1
    const int l15  = lane & 15;

    const int row0 = blockIdx.y * 64;   // M tile base
    const int col0 = blockIdx.x * 128;  // N tile base
    const int m0   = (wave & 3) * 16;
    const int nh   = (wave >> 2) * 64;

    v8f acc[4];
#pragma unroll
    for (int j = 0; j < 4; ++j)
#pragma unroll
        for (int v = 0; v < 8; ++v) acc[j][v] = 0.0f;

    // ---- staging helpers -------------------------------------------------
    auto stageA = [&](int buf, int k0) {
        const int m_idx = tid >> 3;          // 0..31
        const int kq    = (tid & 7) * 4;     // 0,4,...,28
        const float4 g0 = *(const float4*)(X + (size_t)(row0 + m_idx)      * KTOT + k0 + kq);
        const float4 g1 = *(const float4*)(X + (size_t)(row0 + m_idx + 32) * KTOT + k0 + kq);
        Al[buf][kq + 0][m_idx] = g0.x; Al[buf][kq + 1][m_idx] = g0.y;
        Al[buf][kq + 2][m_idx] = g0.z; Al[buf][kq + 3][m_idx] = g0.w;
        Al[buf][kq + 0][m_idx + 32] = g1.x; Al[buf][kq + 1][m_idx + 32] = g1.y;
        Al[buf][kq + 2][m_idx + 32] = g1.z; Al[buf][kq + 3][m_idx + 32] = g1.w;
    };

    auto stageB = [&](int buf, int k0) {
#if HAVE_TDM
        if (wave == 0) {
            // ---- Tensor DMA descriptor: 2D tile 128(x) x 32(k) of 4B elems
            const unsigned long long ga =
                (unsigned long long)(const void*)(W + (size_t)k0 * D_SAE + col0);
            const unsigned lds = (unsigned)(size_t)&Bl[buf][0][0];
            v4u g0;
            g0.x = 1u;                                  // count=1 (valid user D#)
            g0.y = lds;                                 // lds_addr (bytes)
            g0.z = (unsigned)ga;                        // global_addr[31:0]
            g0.w = (unsigned)((ga >> 32) & 0x01FFFFFFull) | (2u << 30); // [56:32] | type=2
            v8i g1;
            // data_size=2 (4B) | pad_enable | pad_interval=6 (128 dw) | pad_amount=15 (16 dw)
            g1[0] = (int)((2u << 16) | (1u << 20) | (6u << 22) | (15u << 25));
            g1[1] = (int)(((unsigned)D_SAE & 0xFFFFu) << 16);  // tensor_dim0[15:0]
            g1[2] = (int)((((unsigned)D_SAE >> 16) & 0xFFFFu)
                        | (((unsigned)KTOT & 0xFFFFu) << 16)); // dim0[31:16] | dim1[15:0]
            g1[3] = (int)((((unsigned)KTOT >> 16) & 0xFFFFu)
                        | (128u << 16));                       // dim1[31:16] | tile_dim0=128
            g1[4] = 32;                                        // tile_dim1=32, tile_dim2=0
            g1[5] = D_SAE;                                     // tensor_dim0_stride lo32
            g1[6] = 0;
            g1[7] = 0;
            const v4i z4 = {0, 0, 0, 0};
#if TDM_SIX_ARGS
            const v8i z8 = {0, 0, 0, 0, 0, 0, 0, 0};
            __builtin_amdgcn_tensor_load_to_lds(g0, g1, z4, z4, z8, 0);
#else
            __builtin_amdgcn_tensor_load_to_lds(g0, g1, z4, z4, 0);
#endif
        }
#elif HAVE_ASYNC
        // every lane copies one 16B chunk per row-group, ASYNCcnt-tracked
        const int c = (tid & 31) * 4;
        const int r = tid >> 5;              // 0..7
#pragma unroll
        for (int rr = 0; rr < 32; rr += 8) {
            gas_ptr g = (gas_ptr)(size_t)(const void*)
                (W + (size_t)(k0 + r + rr) * D_SAE + col0 + c);
            las_ptr l = (las_ptr)(unsigned)(size_t)&Bl[buf][r + rr][c];
            __builtin_amdgcn_global_load_async_to_lds_b128(g, l, 0, 0);
        }
#else
        const int c = (tid & 31) * 4;
        const int r = tid >> 5;              // 0..7
#pragma unroll
        for (int rr = 0; rr < 32; rr += 8) {
            const float4 gb = *(const float4*)(W + (size_t)(k0 + r + rr) * D_SAE + col0 + c);
            *(float4*)&Bl[buf][r + rr][c] = gb;
            if (k0 + 32 < KTOT)
                __builtin_prefetch(W + (size_t)(k0 + 32 + r + rr) * D_SAE + col0 + c, 0, 0);
        }
#endif
    };

    // ---- software pipeline: prefetch tile i+1 while computing tile i -----
    stageB(0, 0);
    stageA(0, 0);

    for (int i = 0; i < NIT; ++i) {
        const int cur = i & 1;
        const int nxt = cur ^ 1;
        const int k0  = i * 32;

        if (i + 1 < NIT) {
            stageB(nxt, k0 + 32);            // DMA: outstanding behind tile i
            stageA(nxt, k0 + 32);
        }
#if HAVE_TDM
        if (wave == 0) {
            if (i + 1 < NIT) __builtin_amdgcn_s_wait_tensorcnt(1);  // tile i done
            else             __builtin_amdgcn_s_wait_tensorcnt(0);  // last tile
        }
#elif HAVE_ASYNC
        // 4 async b128 per wave per tile; in-order completion per wave
        if (i + 1 < NIT) __builtin_amdgcn_s_wait_asynccnt(4);       // tile i done
        else             __builtin_amdgcn_s_wait_asynccnt(0);       // last tile
#endif
        __syncthreads();   // Bl[cur]/Al[cur] visible to all waves

#pragma unroll
        for (int kk = 0; kk < 32; kk += 4) {
            // A fragment: lane = M (0..15), vgpr v holds K = v + 2*half
            v2f a;
            a.x = Al[cur][kk + 2 * half + 0][m0 + l15];
            a.y = Al[cur][kk + 2 * half + 1][m0 + l15];
#pragma unroll
            for (int j = 0; j < 4; ++j) {
                const int n0 = nh + j * 16;
                v2f bf;
                bf.x = Bl[cur][kk + 2 * half + 0][n0 + l15];
                bf.y = Bl[cur][kk + 2 * half + 1][n0 + l15];
                acc[j] = __builtin_amdgcn_wmma_f32_16x16x4_f32(
                    /*neg_a=*/false, a, /*neg_b=*/false, bf,
                    /*c_mod=*/(short)0, acc[j], /*reuse_a=*/false, /*reuse_b=*/false);
            }
        }
        __syncthreads();   // done reading [cur] before iter i+1 overwrites it
    }

    // ---- store: C layout — vgpr v -> row v + 8*half, col = lane&15
#pragma unroll
    for (int j = 0; j < 4; ++j) {
        const int n = col0 + nh + j * 16 + l15;
        const float be = b_enc[n];
#pragma unroll
        for (int v = 0; v < 8; ++v) {
            const int row = row0 + m0 + v + 8 * half;
            pre[(size_t)row * D_SAE + n] = acc[j][v] + be;
        }
    }
}

// ---------------------------------------------------------------------------
// Kernel 2: per-row top-64 + relu, in-place pre -> z, compacted (val,idx) out.
// One block (256 threads) per row; each thread keeps 64 keys in registers.
// ---------------------------------------------------------------------------
__device__ __forceinline__ unsigned ord_key(float f) {
    unsigned u = __float_as_uint(f);
    return (u & 0x80000000u) ? ~u : (u | 0x80000000u);
}
__device__ __forceinline__ float ord_inv(unsigned u) {
    return (u & 0x80000000u) ? __uint_as_float(u ^ 0x80000000u) : __uint_as_float(~u);
}

__global__ __launch_bounds__(256) void topk_kernel(
    float* __restrict__ z,            // in: pre, out: sparse z  (B, D_SAE)
    float* __restrict__ tvals,        // (B, 64)
    int*   __restrict__ tidx)         // (B, 64)
{
    __shared__ unsigned s_cnt;
    __shared__ int s_eq, s_out;

    const int b   = blockIdx.x;
    const int tid = threadIdx.x;
    float* rowp = z + (size_t)b * D_SAE;

    unsigned key[64];
#pragma unroll
    for (int j = 0; j < 64; ++j)
        key[j] = ord_key(rowp[tid + 256 * j]);

    if (tid == 0) { s_eq = 0; s_out = 0; }
    __syncthreads();

    // binary search for the 64th-largest key: largest t with count(>= t) >= 64
    unsigned lo = 0u, hi = 0xFFFFFFFFu;
    for (int it = 0; it < 33 && lo < hi; ++it) {
        const unsigned span = hi - lo;
        const unsigned mid  = lo + (span >> 1) + (span & 1u);   // ceil midpoint
        if (tid == 0) s_cnt = 0;
        __syncthreads();
        unsigned local = 0;
#pragma unroll
        for (int j = 0; j < 64; ++j) local += (key[j] >= mid) ? 1u : 0u;
        atomicAdd(&s_cnt, local);
        __syncthreads();
        const unsigned c = s_cnt;
        __syncthreads();
        if (c >= 64u) lo = mid; else hi = mid - 1u;
    }
    const unsigned thr = lo;

    // count strictly-greater to know how many ties to admit
    if (tid == 0) s_cnt = 0;
    __syncthreads();
    unsigned lgt = 0;
#pragma unroll
    for (int j = 0; j < 64; ++j) lgt += (key[j] > thr) ? 1u : 0u;
    atomicAdd(&s_cnt, lgt);
    __syncthreads();
    const int need_eq = 64 - (int)s_cnt;
    __syncthreads();

    // selection + in-place z write + compaction
#pragma unroll
    for (int j = 0; j < 64; ++j) {
        const int i = tid + 256 * j;
        const unsigned u = key[j];
        bool sel = (u > thr);
        if (!sel && u == thr) {
            const int p = atomicAdd(&s_eq, 1);
            sel = (p < need_eq);
        }
        const float f  = ord_inv(u);
        const float zr = sel ? fmaxf(f, 0.0f) : 0.0f;
        rowp[i] = zr;
        if (sel) {
            const int p = atomicAdd(&s_out, 1);
            tvals[b * 64 + p] = zr;
            tidx [b * 64 + p] = i;
        }
    }
}

// ---------------------------------------------------------------------------
// Kernel 3: sparse decode + basis mix + x_hat + loss.
// One block per row b; thread owns 3 of the 768 output dims.
// ---------------------------------------------------------------------------
__global__ __launch_bounds__(256) void decode_kernel(
    const float* __restrict__ x,       // (B, T, D_IN)
    const float* __restrict__ W_base,  // (KBASIS, D_SAE, D_IN)
    const float* __restrict__ alpha,   // (T, KBASIS)
    const float* __restrict__ b_dec,   // (T, D_IN)
    const float* __restrict__ tvals,   // (B, 64)
    const int*   __restrict__ tidx,    // (B, 64)
    float* __restrict__ x_hat,         // (B, T, D_IN)
    float* __restrict__ loss)          // scalar
{
    __shared__ float sv[64];
    __shared__ int   si[64];
    __shared__ float sal[T_STEPS * KBASIS];
    __shared__ float red[256];

    const int b   = blockIdx.x;
    const int tid = threadIdx.x;

    if (tid < 64) { sv[tid] = tvals[b * 64 + tid]; si[tid] = tidx[b * 64 + tid]; }
    if (tid >= 64 && tid < 64 + T_STEPS * KBASIS) sal[tid - 64] = alpha[tid - 64];
    __syncthreads();

    const int d0 = tid * 3;
    float a0[3] = {0.f, 0.f, 0.f};
    float a1[3] = {0.f, 0.f, 0.f};
    float a2[3] = {0.f, 0.f, 0.f};

    for (int j = 0; j < 64; ++j) {
        const float zv = sv[j];
        if (zv == 0.0f) continue;
        const int s = si[j];
        const float* w0 = W_base + ((size_t)0 * D_SAE + s) * D_IN + d0;
        const float* w1 = W_base + ((size_t)1 * D_SAE + s) * D_IN + d0;
        const float* w2 = W_base + ((size_t)2 * D_SAE + s) * D_IN + d0;
#pragma unroll
        for (int e = 0; e < 3; ++e) {
            a0[e] = fmaf(zv, w0[e], a0[e]);
            a1[e] = fmaf(zv, w1[e], a1[e]);
            a2[e] = fmaf(zv, w2[e], a2[e]);
        }
    }

    float lsum = 0.0f;
#pragma unroll
    for (int t = 0; t < T_STEPS; ++t) {
        const float c0 = sal[t * KBASIS + 0];
        const float c1 = sal[t * KBASIS + 1];
        const float c2 = sal[t * KBASIS + 2];
        const size_t off = ((size_t)b * T_STEPS + t) * D_IN + d0;
#pragma unroll
        for (int e = 0; e < 3; ++e) {
            float xh = c0 * a0[e] + c1 * a1[e] + c2 * a2[e] + b_dec[t * D_IN + d0 + e];
            x_hat[off + e] = xh;
            const float diff = xh - x[off + e];
            lsum = fmaf(diff, diff, lsum);
        }
    }

    red[tid] = lsum;
    __syncthreads();
#pragma unroll
    for (int s2 = 128; s2 > 0; s2 >>= 1) {
        if (tid < s2) red[tid] += red[tid + s2];
        __syncthreads();
    }
    if (tid == 0)
        atomicAdd(loss, red[0] * (1.0f / (float)(BATCH * T_STEPS)));
}

// ---------------------------------------------------------------------------
// Host launcher
// Inputs (setup_inputs order): x, W_enc, b_enc, W_base, alpha, b_dec, k
// Output: [loss (1)] [x_hat (B*T*D_IN)] [z (B*D_SAE)]
// ---------------------------------------------------------------------------
extern "C" void kernel_launch(void* const* d_in, const int* in_sizes, int n_in,
                              void* d_out, int out_size, void* d_ws, size_t ws_size,
                              hipStream_t stream) {
    const float* x      = (const float*)d_in[0];
    const float* W_enc  = (const float*)d_in[1];
    const float* b_enc  = (const float*)d_in[2];
    const float* W_base = (const float*)d_in[3];
    const float* alpha  = (const float*)d_in[4];
    const float* b_dec  = (const float*)d_in[5];
    (void)in_sizes; (void)n_in; (void)out_size; (void)ws_size;

    float* out   = (float*)d_out;
    float* x_hat = out + 1;
    float* zbuf  = out + 1 + (size_t)BATCH * T_STEPS * D_IN;  // z region; doubles as 'pre'

    float* tvals = (float*)d_ws;                              // B*64 floats
    int*   tidx  = (int*)((char*)d_ws + (size_t)BATCH * 64 * sizeof(float));

    // zero the loss accumulator (graph-capture legal)
    hipMemsetAsync(d_out, 0, sizeof(float), stream);

    // 1) encode GEMM -> pre (stored in z region)
    encode_gemm_kernel<<<dim3(D_SAE / 128, BATCH / 64), 256, 0, stream>>>(
        x, W_enc, b_enc, zbuf);

    // 2) top-64 + relu, in-place pre -> z, compacted lists to workspace
    topk_kernel<<<BATCH, 256, 0, stream>>>(zbuf, tvals, tidx);

    // 3) sparse decode + x_hat + loss
    decode_kernel<<<BATCH, 256, 0, stream>>>(
        x, W_base, alpha, b_dec, tvals, tidx, x_hat, out);
}